// MultiSpark_19997367730510
// MI455X (gfx1250) — compile-verified
//
#include <hip/hip_runtime.h>
#include <math.h>

// Problem constants (match reference)
#define NN 8192
#define KK 64
#define KNEIGH 5
#define MAX_OVR (KK * 36)   // 1 hebbian + 5 + 5 + 25 ripple ops per step

typedef __attribute__((ext_vector_type(2))) float v2f;
typedef __attribute__((ext_vector_type(8))) float v8f;

__device__ __forceinline__ float decay_clip(float w) {
    float v = w * 0.999f;               // 1 - LR_GLOBAL_DECAY
    v = fminf(v, 2.0f);
    v = fmaxf(v, -2.0f);
    return v;
}

// ---------------------------------------------------------------------------
// Kernel A: one fused streaming pass over W (read once, 256 MB):
//   - accumulate y = W @ (0.95*s) with V_WMMA_F32_16X16X4_F32
//   - write Wout = clip(0.999*W, +-2) on the way through
//   - s_mid = sigmoid(y + 0.05*noise)
// 128 threads = 4 waves; each wave owns 16 consecutive rows.
// ---------------------------------------------------------------------------
__global__ __launch_bounds__(128)
void matvec_decay_kernel(const float* __restrict__ W,
                         const float* __restrict__ s,
                         const float* __restrict__ noise,
                         float* __restrict__ Wout,
                         float* __restrict__ s_mid) {
    __shared__ float s_lds[NN];          // 0.95*s, broadcast-read by all waves
    __shared__ float tile[4][16 * 68];   // per-wave 16x64 tile, stride 68 (16B aligned, bank-safe)

    const int tid  = threadIdx.x;
    const int wave = tid >> 5;
    const int lane = tid & 31;
    const int rows = (blockIdx.x * 4 + wave) * 16;

    for (int j = tid; j < NN; j += 128) s_lds[j] = 0.95f * s[j];   // STATE_DECAY
    __syncthreads();

    const int m    = lane & 15;          // A-matrix row within tile (ISA 16x4 f32 layout)
    const int koff = (lane >> 4) * 2;    // lanes 16-31 hold K=2,3
    float* tw = tile[wave];

    v8f acc = {};                        // 16x16 f32 C/D accumulator (8 VGPRs)

    for (int kc = 0; kc < NN; kc += 64) {
        // Stage 16x64 tile (coalesced 256B row segments) + decayed store.
        #pragma unroll
        for (int i = 0; i < 8; ++i) {
            int t = lane * 4 + i * 128;  // flat float index in 16x64 tile
            int r = t >> 6;
            int c = t & 63;
            size_t off = (size_t)(rows + r) * NN + kc + c;
            const float4 v = *(const float4*)(W + off);
            float4 d;
            d.x = decay_clip(v.x); d.y = decay_clip(v.y);
            d.z = decay_clip(v.z); d.w = decay_clip(v.w);
            *(float4*)(Wout + off) = d;
            *(float4*)&tw[r * 68 + c] = v;
            __builtin_prefetch((const char*)(W + off + 64), 0, 1);  // next k-chunk
        }
        __syncthreads();

        // 16 WMMAs per tile: A = 16x4 W chunk, B = 0.95*s[k..k+3] broadcast
        // into all 16 columns => every column of C accumulates y[m].
        #pragma unroll
        for (int kk = 0; kk < 64; kk += 4) {
            v2f a, b;
            a.x = tw[m * 68 + kk + koff];
            a.y = tw[m * 68 + kk + koff + 1];
            b.x = s_lds[kc + kk + koff];
            b.y = s_lds[kc + kk + koff + 1];
            acc = __builtin_amdgcn_wmma_f32_16x16x4_f32(
                false, a, false, b, (short)0, acc, false, false);
        }
        __syncthreads();
    }

    // C layout: VGPR j, lanes 0-15 -> M=j, lanes 16-31 -> M=8+j (all columns equal).
    if (lane == 0 || lane == 16) {
        int mb = rows + ((lane >> 4) ? 8 : 0);
        #pragma unroll
        for (int j = 0; j < 8; ++j) {
            float y = acc[j] + 0.05f * noise[mb + j];       // NOISE_STD
            s_mid[mb + j] = 1.0f / (1.0f + expf(-y));       // sigmoid
        }
    }
}

// ---------------------------------------------------------------------------
// Kernel B: sequential 64-step scan in ONE block (1024 threads).
// Sparse W updates kept as an LDS overlay log; each step materializes the
// effective row W[prev] (base row + overlay) in LDS, does argmax / top-5
// with tie->lowest-index reductions, then patches touched entries of the
// already-decayed out_W at the end.
// Note: mode 1's scalar Gumbel term cannot change an argmax, so modes 0/1
// are both plain argmax(relu(row)+1e-6); `unif` is legitimately unused.
// ---------------------------------------------------------------------------
__global__ __launch_bounds__(1024)
void scan_kernel(const float* __restrict__ W,
                 float* __restrict__ s_ws,          // s_mid in d_ws, updated in place
                 const float* __restrict__ spark_energy,
                 const int*   __restrict__ spark_pos,
                 const int*   __restrict__ spark_age,
                 const int*   __restrict__ randint_vals,
                 const int*   __restrict__ step_num,
                 int*   __restrict__ out_pos,
                 float* __restrict__ out_s,
                 float* __restrict__ out_W,
                 float* __restrict__ out_energy,
                 int*   __restrict__ out_age) {
    __shared__ float row_buf[NN];        // 32 KB effective row
    __shared__ int   ovr_rc[MAX_OVR];    // (r<<13)|c, bit30 = hebbian
    __shared__ float ovr_val[MAX_OVR];
    __shared__ float rv[1024];
    __shared__ int   ri[1024];
    __shared__ int   next_all[KK];
    __shared__ int   top5[KNEIGH];
    __shared__ int   novr;
    __shared__ int   np_sh;

    const int tid = threadIdx.x;
    if (tid == 0) novr = 0;

    // Force young sparks to 1.0 (spark_pos always in range; old sparks dropped).
    if (tid < KK && spark_age[tid] < 5) s_ws[spark_pos[tid]] = 1.0f;

    const int mode = step_num[0] % 3;    // uniform -> no divergent barriers
    __syncthreads();

    for (int i = 0; i < KK; ++i) {
        const int prev    = spark_pos[i];
        const float e_dec = spark_energy[i] * 0.98f;   // SPARK_ENERGY_DECAY
        const int nov     = novr;

        // Materialize effective row: base + overlay entries touching row `prev`.
        for (int j = tid; j < NN; j += 1024) row_buf[j] = W[(size_t)prev * NN + j];
        __syncthreads();
        for (int e = tid; e < nov; e += 1024) {
            int rc = ovr_rc[e];
            if (((rc >> 13) & (NN - 1)) == prev) {
                int c = rc & (NN - 1);
                if (rc & (1 << 30))
                    row_buf[c] = row_buf[c] * 0.95f + ovr_val[e] * 0.05f; // hebbian (rare)
                else
                    atomicAdd(&row_buf[c], ovr_val[e]);
            }
        }
        __syncthreads();

        int np;
        if (mode == 2) {
            np = randint_vals[i];
        } else {
            // argmax(relu(row)+1e-6), ties -> lowest index (blocked partition).
            float bv = -1.0f; int bi = 0;
            int j0 = tid * (NN / 1024);
            #pragma unroll
            for (int j = 0; j < NN / 1024; ++j) {
                float v = row_buf[j0 + j];
                v = (v > 0.0f ? v : 0.0f) + 1e-6f;
                if (v > bv) { bv = v; bi = j0 + j; }
            }
            rv[tid] = bv; ri[tid] = bi;
            __syncthreads();
            for (int off = 512; off > 0; off >>= 1) {
                if (tid < off) {
                    float ov = rv[tid + off]; int oi = ri[tid + off];
                    if (ov > rv[tid] || (ov == rv[tid] && oi < ri[tid])) {
                        rv[tid] = ov; ri[tid] = oi;
                    }
                }
                __syncthreads();
            }
            np = ri[0];
        }

        if (tid == 0) {
            np_sh = np;
            next_all[i] = np;
            // Hebbian: W[np,prev] = W[np,prev]*0.95 + s[prev]*0.05 (effective value)
            float base;
            if (np == prev) {
                base = row_buf[prev];
            } else {
                base = W[(size_t)np * NN + prev];
                for (int e = 0; e < nov; ++e) {
                    int rc = ovr_rc[e];
                    if (((rc >> 13) & (NN - 1)) == np && (rc & (NN - 1)) == prev) {
                        if (rc & (1 << 30)) base = base * 0.95f + ovr_val[e] * 0.05f;
                        else                base += ovr_val[e];
                    }
                }
            }
            float s_prev = s_ws[prev];
            int n0 = novr;
            ovr_rc[n0]  = (np << 13) | prev | (1 << 30);
            ovr_val[n0] = s_prev;
            novr = n0 + 1;
            if (np == prev) row_buf[prev] = base * 0.95f + s_prev * 0.05f;
            s_ws[np] = e_dec;            // after reading s[prev], per reference order
        }
        __syncthreads();
        np = np_sh;

        // top-5 of relu(row) (post-hebbian), ties -> lowest index
        for (int r5 = 0; r5 < KNEIGH; ++r5) {
            float bv = -1.0f; int bi = 0;
            int j0 = tid * (NN / 1024);
            for (int j = 0; j < NN / 1024; ++j) {
                int idx = j0 + j;
                float v = row_buf[idx];
                v = v > 0.0f ? v : 0.0f;
                bool skip = false;
                for (int q = 0; q < r5; ++q) skip = skip || (top5[q] == idx);
                if (!skip && v > bv) { bv = v; bi = idx; }
            }
            rv[tid] = bv; ri[tid] = bi;
            __syncthreads();
            for (int off = 512; off > 0; off >>= 1) {
                if (tid < off) {
                    float ov = rv[tid + off]; int oi = ri[tid + off];
                    if (ov > rv[tid] || (ov == rv[tid] && oi < ri[tid])) {
                        rv[tid] = ov; ri[tid] = oi;
                    }
                }
                __syncthreads();
            }
            if (tid == 0) top5[r5] = ri[0];
            __syncthreads();
        }

        // Ripple: append 5 + 5 + 25 = 35 add-ops to the overlay.
        {
            int n0 = novr;               // uniform after barrier
            if (tid < 35) {
                int r, c; float v;
                if (tid < 5)       { r = prev;          c = top5[tid];     v = 0.01f;  }
                else if (tid < 10) { r = top5[tid - 5]; c = prev;          v = 0.005f; }
                else { int t = tid - 10; r = top5[t / 5]; c = top5[t % 5]; v = 0.003f; }
                ovr_rc[n0 + tid]  = (r << 13) | c;
                ovr_val[n0 + tid] = v;
            }
            __syncthreads();
            if (tid == 0) novr = n0 + 35;
            __syncthreads();
        }
    }

    // Spark outputs (base_pos = i % N = i since K <= N).
    if (tid < KK) {
        float e_dec = spark_energy[tid] * 0.98f;
        bool reset  = e_dec < 0.05f;     // SPARK_MIN_ENERGY
        out_pos[tid]    = reset ? tid : next_all[tid];
        out_energy[tid] = reset ? 1.0f : e_dec;
        out_age[tid]    = reset ? 0 : spark_age[tid] + 1;
    }
    for (int j = tid; j < NN; j += 1024) out_s[j] = s_ws[j];
    __syncthreads();

    // Patch touched W entries: recompute each from base with full op-ordered
    // overlay, then decay+clip. Duplicate writes produce identical values.
    const int tot = novr;
    for (int e = tid; e < tot; e += 1024) {
        int rc = ovr_rc[e] & ~(1 << 30);
        int r = (rc >> 13) & (NN - 1), c = rc & (NN - 1);
        float v = W[(size_t)r * NN + c];
        for (int q = 0; q < tot; ++q) {
            int qrc = ovr_rc[q];
            if ((qrc & ~(1 << 30)) == rc) {
                if (qrc & (1 << 30)) v = v * 0.95f + ovr_val[q] * 0.05f;
                else                 v += ovr_val[q];
            }
        }
        out_W[(size_t)r * NN + c] = decay_clip(v);
    }
}

// ---------------------------------------------------------------------------
extern "C" void kernel_launch(void* const* d_in, const int* in_sizes, int n_in,
                              void* d_out, int out_size, void* d_ws, size_t ws_size,
                              hipStream_t stream) {
    const float* W            = (const float*)d_in[0];
    const float* s            = (const float*)d_in[1];
    const float* noise        = (const float*)d_in[2];
    // d_in[3] = unif: unused — per-step Gumbel noise is a scalar, so it cannot
    // change the argmax (mode 1 == mode 0).
    const float* spark_energy = (const float*)d_in[4];
    const int*   spark_pos    = (const int*)d_in[5];
    const int*   spark_age    = (const int*)d_in[6];
    const int*   randint_vals = (const int*)d_in[7];
    const int*   step_num     = (const int*)d_in[8];

    // Outputs concatenated flat in return order: pos(K i32), s(N f32),
    // W(N*N f32), energy(K f32), age(K i32).
    float* out     = (float*)d_out;
    int*   out_pos = (int*)out;
    float* out_s   = out + KK;
    float* out_W   = out + KK + NN;
    float* out_e   = out + KK + NN + (size_t)NN * NN;
    int*   out_age = (int*)(out + KK + NN + (size_t)NN * NN + KK);

    float* s_mid = (float*)d_ws;   // [N] intermediate sigmoid state

    matvec_decay_kernel<<<NN / 64, 128, 0, stream>>>(W, s, noise, out_W, s_mid);
    scan_kernel<<<1, 1024, 0, stream>>>(W, s_mid, spark_energy, spark_pos, spark_age,
                                        randint_vals, step_num,
                                        out_pos, out_s, out_W, out_e, out_age);
}